// CrossHeadAttention_82789789598473
// MI455X (gfx1250) — compile-verified
//
#include <hip/hip_runtime.h>
#include <math.h>

typedef __attribute__((ext_vector_type(2))) float v2f;
typedef __attribute__((ext_vector_type(8))) float v8f;

#define LN_EPS 1e-5f
#define ATTN_SCALE 0.35355339059327373f   // 8^-0.5
#define PLANE_F4 16384                    // 65536 floats / 4 per plane

// ---------------------------------------------------------------------------
// Kernel A: spatial mean pool. One block per (b,h,c) plane of 256*256 floats.
// 256 threads * 64 float4 each = 65536 floats. Pure HBM streaming read.
// ---------------------------------------------------------------------------
__global__ __launch_bounds__(256) void pool_kernel(const float* __restrict__ x,
                                                   float* __restrict__ pooled) {
    const int plane = blockIdx.x;                       // 0..1023
    const float4* x4 = (const float4*)x + (size_t)plane * PLANE_F4;
    const int t = threadIdx.x;
    float s = 0.f;
    #pragma unroll 8
    for (int j = 0; j < 64; ++j) {
        float4 v = x4[t + j * 256];
        s += (v.x + v.y) + (v.z + v.w);
    }
    __shared__ float red[256];
    red[t] = s;
    __syncthreads();
    for (int off = 128; off > 0; off >>= 1) {
        if (t < off) red[t] += red[t + off];
        __syncthreads();
    }
    if (t == 0) pooled[plane] = red[0] * (1.0f / 65536.0f);
}

// ---------------------------------------------------------------------------
// Kernel B: the tiny transformer math on pooled (32,4,8), one block of 128
// threads (4 waves), thread r = row (b*4+h).  Q/K/V projections done with
// V_WMMA_F32_16X16X4_F32 (two K=4 steps cover CH=8); everything else scalar.
// Emits awp1 = 1 + (sigmoid(gate)*(x_out - x_pooled) + x_pooled), 1024 floats.
// ---------------------------------------------------------------------------
__global__ __launch_bounds__(128) void tiny_attn_kernel(
    const float* __restrict__ pooled,
    const float* __restrict__ Wq, const float* __restrict__ Wk,
    const float* __restrict__ Wv, const float* __restrict__ Wo,
    const float* __restrict__ bo,
    const float* __restrict__ W1, const float* __restrict__ b1,
    const float* __restrict__ W2, const float* __restrict__ b2,
    const float* __restrict__ g1, const float* __restrict__ beta1,
    const float* __restrict__ g2, const float* __restrict__ beta2,
    const float* __restrict__ gate,
    float* __restrict__ awp1)
{
    __shared__ float xp[128][8];   // pooled rows
    __shared__ float xn[128][8];   // LN1 output
    __shared__ float Qs[128][8];
    __shared__ float Ks[128][8];
    __shared__ float Vs[128][8];

    const int r = threadIdx.x;     // row = b*4 + h

    // ---- phase 1: load pooled row, LayerNorm 1 ----
    float v[8];
    float mu = 0.f;
    #pragma unroll
    for (int c = 0; c < 8; ++c) { v[c] = pooled[r * 8 + c]; mu += v[c]; }
    mu *= 0.125f;
    float var = 0.f;
    #pragma unroll
    for (int c = 0; c < 8; ++c) { float d = v[c] - mu; var += d * d; }
    var *= 0.125f;
    float inv = rsqrtf(var + LN_EPS);
    #pragma unroll
    for (int c = 0; c < 8; ++c) {
        xp[r][c] = v[c];
        xn[r][c] = (v[c] - mu) * inv * g1[c] + beta1[c];
    }
    __syncthreads();

    // ---- phase 2: Q/K/V = xn(128x8) @ W^T(8x8) via WMMA f32 16x16x4 ----
    const int lane  = threadIdx.x & 31;
    const int wave  = threadIdx.x >> 5;
    const int nEff  = lane & 15;          // row (A) / col (B,D) within tile
    const int khalf = (lane >> 4) << 1;   // 0 for lanes 0-15, 2 for 16-31
    const float colMask = (nEff < 8) ? 1.0f : 0.0f;
    const int   nClamp  = (nEff < 8) ? nEff : 0;

    const float* Wmats[3] = {Wq, Wk, Wv};
    float (*Dsts[3])[8]   = {Qs, Ks, Vs};

    #pragma unroll
    for (int tIdx = 0; tIdx < 2; ++tIdx) {
        const int base = (wave * 2 + tIdx) * 16;
        const int rowA = base + nEff;
        // A fragments (16x4 layout: VGPR0/1 = K pair, lane>=16 holds K+2)
        v2f a0, a1;
        a0[0] = xn[rowA][khalf];     a0[1] = xn[rowA][khalf + 1];
        a1[0] = xn[rowA][4 + khalf]; a1[1] = xn[rowA][4 + khalf + 1];
        #pragma unroll
        for (int m = 0; m < 3; ++m) {
            const float* W = Wmats[m];
            // B[k][n] = W[n][k]; columns 8..15 zero-padded
            v2f b0, b1f;
            b0[0]  = W[nClamp * 8 + khalf]         * colMask;
            b0[1]  = W[nClamp * 8 + khalf + 1]     * colMask;
            b1f[0] = W[nClamp * 8 + 4 + khalf]     * colMask;
            b1f[1] = W[nClamp * 8 + 4 + khalf + 1] * colMask;
            v8f c = {};
            c = __builtin_amdgcn_wmma_f32_16x16x4_f32(false, a0, false, b0,
                                                      (short)0, c, false, false);
            c = __builtin_amdgcn_wmma_f32_16x16x4_f32(false, a1, false, b1f,
                                                      (short)0, c, false, false);
            // D scatter: VGPR j, lanes 0-15 -> M=j, lanes 16-31 -> M=j+8; N=lane&15
            if (nEff < 8) {
                const int mb = base + ((lane >> 4) << 3);
                #pragma unroll
                for (int j = 0; j < 8; ++j) Dsts[m][mb + j][nEff] = c[j];
            }
        }
    }
    __syncthreads();

    // ---- phase 3: attention across the 4 heads of this row's batch ----
    const int bRow = (r >> 2) << 2;       // first row of this batch
    float q[8];
    #pragma unroll
    for (int c = 0; c < 8; ++c) q[c] = Qs[r][c];
    float sc[4], mx = -1e30f;
    #pragma unroll
    for (int g = 0; g < 4; ++g) {
        float acc = 0.f;
        #pragma unroll
        for (int c = 0; c < 8; ++c) acc += q[c] * Ks[bRow + g][c];
        sc[g] = acc * ATTN_SCALE;
        mx = fmaxf(mx, sc[g]);
    }
    float se = 0.f;
    #pragma unroll
    for (int g = 0; g < 4; ++g) { sc[g] = expf(sc[g] - mx); se += sc[g]; }
    const float rs = 1.0f / se;

    float ao[8];
    #pragma unroll
    for (int c = 0; c < 8; ++c) {
        float acc = 0.f;
        #pragma unroll
        for (int g = 0; g < 4; ++g) acc += sc[g] * Vs[bRow + g][c];
        ao[c] = acc * rs;
    }
    float xa[8];
    #pragma unroll
    for (int c = 0; c < 8; ++c) {
        float acc = bo[c];
        #pragma unroll
        for (int k = 0; k < 8; ++k) acc += ao[k] * Wo[c * 8 + k];
        xa[c] = xp[r][c] + acc;
    }

    // ---- LayerNorm 2 + exact-GELU FFN ----
    mu = 0.f;
    #pragma unroll
    for (int c = 0; c < 8; ++c) mu += xa[c];
    mu *= 0.125f;
    var = 0.f;
    #pragma unroll
    for (int c = 0; c < 8; ++c) { float d = xa[c] - mu; var += d * d; }
    var *= 0.125f;
    inv = rsqrtf(var + LN_EPS);
    float xn2[8];
    #pragma unroll
    for (int c = 0; c < 8; ++c) xn2[c] = (xa[c] - mu) * inv * g2[c] + beta2[c];

    float h1[4];
    #pragma unroll
    for (int j = 0; j < 4; ++j) {
        float acc = b1[j];
        #pragma unroll
        for (int c = 0; c < 8; ++c) acc += xn2[c] * W1[j * 8 + c];
        h1[j] = 0.5f * acc * (1.0f + erff(acc * 0.70710678118654752f));
    }
    const float gt = 1.0f / (1.0f + expf(-gate[0]));
    #pragma unroll
    for (int c = 0; c < 8; ++c) {
        float acc = b2[c];
        #pragma unroll
        for (int j = 0; j < 4; ++j) acc += h1[j] * W2[c * 4 + j];
        const float xo  = xa[c] + acc;
        const float xpv = xp[r][c];
        awp1[r * 8 + c] = 1.0f + (gt * (xo - xpv) + xpv);
    }
}

// ---------------------------------------------------------------------------
// Kernel C: out = x * awp1[plane].  4 blocks per plane; scale is block-uniform
// (plane = blockIdx>>2) so the awp1 load becomes an SMEM scalar load. b128
// vectorized, stride-256 coalesced streaming: the HBM-bound 512 MB pass.
// ---------------------------------------------------------------------------
__global__ __launch_bounds__(256) void scale_kernel(const float* __restrict__ x,
                                                    const float* __restrict__ awp1,
                                                    float* __restrict__ out) {
    const int plane = blockIdx.x >> 2;      // uniform across block
    const int chunk = blockIdx.x & 3;
    const float s = awp1[plane];
    const size_t base = (size_t)plane * PLANE_F4 + chunk * 4096 + threadIdx.x;
    const float4* x4 = (const float4*)x;
    float4* o4 = (float4*)out;
    #pragma unroll
    for (int j = 0; j < 16; ++j) {
        const size_t i = base + (size_t)j * 256;
        float4 vv = x4[i];
        vv.x *= s; vv.y *= s; vv.z *= s; vv.w *= s;
        o4[i] = vv;
    }
}

extern "C" void kernel_launch(void* const* d_in, const int* in_sizes, int n_in,
                              void* d_out, int out_size, void* d_ws, size_t ws_size,
                              hipStream_t stream) {
    const float* x     = (const float*)d_in[0];
    const float* Wq    = (const float*)d_in[1];
    const float* Wk    = (const float*)d_in[2];
    const float* Wv    = (const float*)d_in[3];
    const float* Wo    = (const float*)d_in[4];
    const float* bo    = (const float*)d_in[5];
    const float* W1    = (const float*)d_in[6];
    const float* b1    = (const float*)d_in[7];
    const float* W2    = (const float*)d_in[8];
    const float* b2    = (const float*)d_in[9];
    const float* g1    = (const float*)d_in[10];
    const float* beta1 = (const float*)d_in[11];
    const float* g2    = (const float*)d_in[12];
    const float* beta2 = (const float*)d_in[13];
    const float* gate  = (const float*)d_in[14];
    float* out = (float*)d_out;

    float* pooled = (float*)d_ws;        // 1024 floats
    float* awp1   = pooled + 1024;       // 1024 floats

    pool_kernel<<<1024, 256, 0, stream>>>(x, pooled);
    tiny_attn_kernel<<<1, 128, 0, stream>>>(pooled, Wq, Wk, Wv, Wo, bo,
                                            W1, b1, W2, b2, g1, beta1,
                                            g2, beta2, gate, awp1);
    scale_kernel<<<4096, 256, 0, stream>>>(x, awp1, out);
}